// KalmanMeasNoise_62646392980134
// MI455X (gfx1250) — compile-verified
//
#include <hip/hip_runtime.h>

// MI455X / gfx1250, wave32. B=8192, 16 elements per wave with 2x redundant
// ownership (lanes j and 16+j both carry element base+j) -> 512 waves for
// latency hiding, and a shuffle-free WMMA B-matrix: each lane feeds its own
// u/v (low half, K=0/1) or dv (high half, K=2) into V_WMMA_F32_16X16X4_F32.
// A = W1^T (hidden on M, constant regs). Layer-2 = 16 per-lane FMAs (4-way
// split accumulators) + one v_permlanex16 (lane^16 half-sum, pure VALU).

#define BATCH 8192
#define LHIST 512
#define HFUT  64

typedef __attribute__((ext_vector_type(2))) float v2f;
typedef __attribute__((ext_vector_type(8))) float v8f;

#if __has_builtin(__builtin_amdgcn_wmma_f32_16x16x4_f32)
#define HAVE_WMMA_F32X4 1
#else
#define HAVE_WMMA_F32X4 0
#endif

#define LOG2E 1.44269504088896340736f

__device__ __forceinline__ float fast_exp2f(float x) {
#if __has_builtin(__builtin_amdgcn_exp2f)
  return __builtin_amdgcn_exp2f(x);          // v_exp_f32, single TRANS op
#else
  return exp2f(x);
#endif
}

__device__ __forceinline__ float fast_rcpf(float x) {
#if __has_builtin(__builtin_amdgcn_rcpf)
  return __builtin_amdgcn_rcpf(x);           // v_rcp_f32: no div_scale/fixup chain
#else
  return 1.0f / x;
#endif
}

__device__ __forceinline__ float fast_tanhf(float x) {
#if __has_builtin(__builtin_amdgcn_tanhf)
  return __builtin_amdgcn_tanhf(x);          // v_tanh_f32 (TRANS pipe, co-executes)
#else
  float e = fast_exp2f(2.0f * LOG2E * x);
  return 1.0f - 2.0f * fast_rcpf(e + 1.0f);
#endif
}

// x[lane] + x[lane^16], computed on the VALU (v_permlanex16_b32, identity
// selects) instead of ds_bpermute, keeping the serial chain off the LDS pipe.
__device__ __forceinline__ float xor16_sum(float x) {
#if __has_builtin(__builtin_amdgcn_permlanex16)
  int o = __builtin_amdgcn_permlanex16(__float_as_int(x), __float_as_int(x),
                                       0x76543210, 0xFEDCBA98, false, false);
  return x + __int_as_float(o);
#else
  return x + __shfl_xor(x, 16, 32);
#endif
}

struct KArgs {
  const float *v_hist, *dt_hist, *x_obs, *v_fut, *dt_fut;
  const float *log_tau_n, *log_q_n, *log_R_white, *log_P0_nn, *log_p0_xx, *log_p0_uu;
  const float *alpha, *c, *kappa, *vc, *qx, *qu, *q_scale;
  const float *W1, *b1, *W2, *b2;
  float *out;
};

__global__ __launch_bounds__(32) void kf_neural_closure_kernel(KArgs A) {
  const int  lane = threadIdx.x;
  const bool lo   = lane < 16;
  const int  j    = lane & 15;
  const int  e    = blockIdx.x * 16 + j;      // element co-owned by lanes j, 16+j
  const int  hoff = lo ? 0 : 8;               // C-layout: halves hold M=r / M=r+8

  // ---------------- uniform scalar parameters ----------------
  const float alpha   = A.alpha[0];
  const float cpar    = A.c[0];
  const float kappa   = A.kappa[0];
  const float vcv     = A.vc[0];
  const float vc2     = vcv * vcv;
  const float inv_tau = fast_exp2f(-LOG2E * A.log_tau_n[0]);   // 1/tau_n
  const float q_n     = fast_exp2f(LOG2E * A.log_q_n[0]);
  const float R_w     = fast_exp2f(LOG2E * A.log_R_white[0]);
  const float qxs     = A.q_scale[0] * A.qx[0];
  const float qus     = A.q_scale[0] * A.qu[0];
  const float b2s     = A.b2[0];
  const float c_rho   = -alpha * LOG2E;       // rho = exp2(dt * c_rho)
  const float c_phi   = -inv_tau * LOG2E;     // phi = exp2(dt * c_phi)

  // ---------------- WMMA-resident MLP constants ----------------
  // A-matrix 16x4 f32 layout: lanes0-15 (M=lane): v.x=K0, v.y=K1
  //                           lanes16-31 (M=lane-16): v.x=K2, v.y=K3(pad=0)
  // A tile a holds W1^T rows (hidden 16a..16a+15), K = input feature.
  v2f Am0, Am1;
  Am0.x = lo ? A.W1[0 * 32 + j]      : A.W1[2 * 32 + j];
  Am0.y = lo ? A.W1[1 * 32 + j]      : 0.0f;
  Am1.x = lo ? A.W1[0 * 32 + 16 + j] : A.W1[2 * 32 + 16 + j];
  Am1.y = lo ? A.W1[1 * 32 + 16 + j] : 0.0f;

  // C/D layout: (VGPR r, lane) -> M = r + 8*(lane>=16), N = lane&15.
  // Bias enters through the C operand; W2 weights become per-lane scalars.
  v8f cb0, cb1;
  float w2c0[8], w2c1[8];
#pragma unroll
  for (int r = 0; r < 8; ++r) {
    cb0[r]  = A.b1[hoff + r];
    cb1[r]  = A.b1[16 + hoff + r];
    w2c0[r] = A.W2[hoff + r];
    w2c1[r] = A.W2[16 + hoff + r];
  }

  // closure(u,v,dv): wave-wide MLP; both co-owner lanes receive the result.
  auto closure = [&](float u, float v, float dv) -> float {
#if HAVE_WMMA_F32X4
    // B-matrix 4x16 (K x N=elem): every lane supplies its OWN registers.
    v2f Bm;
    Bm.x = lo ? u : dv;                       // K0 row (u) | K2 row (dv)
    Bm.y = lo ? v : 0.0f;                     // K1 row (v) | K3 pad
    v8f h0 = __builtin_amdgcn_wmma_f32_16x16x4_f32(false, Am0, false, Bm, (short)0, cb0, false, false);
    v8f h1 = __builtin_amdgcn_wmma_f32_16x16x4_f32(false, Am1, false, Bm, (short)0, cb1, false, false);
    // 4-way split accumulators: ~4-deep dependent FMA chain instead of 16-deep.
    float pa = 0.0f, pb = 0.0f, pc = 0.0f, pd = 0.0f;
#pragma unroll
    for (int r = 0; r < 8; r += 2) {
      pa = fmaf(fast_tanhf(h0[r]),     w2c0[r],     pa);
      pb = fmaf(fast_tanhf(h0[r + 1]), w2c0[r + 1], pb);
      pc = fmaf(fast_tanhf(h1[r]),     w2c1[r],     pc);
      pd = fmaf(fast_tanhf(h1[r + 1]), w2c1[r + 1], pd);
    }
    float p = (pa + pb) + (pc + pd);
    return xor16_sum(p) + b2s;                // combine halves; broadcast to both
#else
    float acc = b2s;
    for (int h = 0; h < 32; ++h) {
      float pre = fmaf(u, A.W1[h], fmaf(v, A.W1[32 + h], fmaf(dv, A.W1[64 + h], A.b1[h])));
      acc = fmaf(fast_tanhf(pre), A.W2[h], acc);
    }
    return acc;
#endif
  };

  // ---------------- per-lane (co-owned) state ----------------
  const float* rowV   = A.v_hist  + (size_t)e * LHIST;
  const float* rowDT  = A.dt_hist + (size_t)e * LHIST;
  const float* rowOBS = A.x_obs   + (size_t)e * LHIST;

  float s0 = rowOBS[0], s1 = 0.0f, s2 = 0.0f;
  float p00 = fast_exp2f(LOG2E * A.log_p0_xx[0]);
  float p11 = fast_exp2f(LOG2E * A.log_p0_uu[0]);
  float p22 = fast_exp2f(LOG2E * A.log_P0_nn[0]);
  float p01 = 0.0f, p02 = 0.0f, p12 = 0.0f;

  // predict: state + covariance time update (F has structural zeros)
  auto predict = [&](float dt_in, float vcur, float dv) {
    float dt  = fmaxf(dt_in, 1e-6f);
    float rho = fast_exp2f(dt * c_rho);       // exp(-alpha*dt)
    float phi = fast_exp2f(dt * c_phi);       // exp(-dt/tau_n)
    float g   = fmaxf(fmaf(vcur, vcur, -vc2), 0.0f);
    float cl  = closure(s1, vcur, dv);
    float a   = -kappa * dt;
    float x_p = fmaf(s1, dt, s0);
    float u_p = rho * s1 + a * s0 + (cpar * g + cl) * dt;
    float n_p = phi * s2;
    float t0 = fmaf(dt, p01, p00);
    float t1 = fmaf(dt, p11, p01);
    float t2 = fmaf(dt, p12, p02);
    float r0 = a * p00 + rho * p01;
    float r1 = a * p01 + rho * p11;
    float r2 = a * p02 + rho * p12;
    float np00 = t0 + dt * t1 + qxs * dt;
    float np01 = a * t0 + rho * t1;
    float np02 = phi * t2;
    float np11 = a * r0 + rho * r1 + qus * dt;
    float np12 = phi * r2;
    float np22 = phi * phi * p22 + q_n;
    s0 = x_p; s1 = u_p; s2 = n_p;
    p00 = np00; p01 = np01; p02 = np02; p11 = np11; p12 = np12; p22 = np22;
  };

  // ---------------- filter scan: k = 1..L-1 ----------------
  float vB = rowV[0];
#pragma unroll 2
  for (int k = 1; k < LHIST; ++k) {
    if ((k & 15) == 1) {                      // stream history ahead of use
      int kp = (k + 64 < LHIST) ? k + 64 : LHIST - 1;
      __builtin_prefetch(rowDT + kp, 0, 1);
      __builtin_prefetch(rowOBS + kp, 0, 1);
      __builtin_prefetch(rowV + kp, 0, 1);
    }
    float dtk = rowDT[k];
    float obs = rowOBS[k];
    float vA  = rowV[k - 1];
    float dv  = vA - vB;                      // k==1: vA==vB -> dv=0
    vB = vA;

    predict(dtk, vA, dv);

    // measurement update, H = [1,0,1], Joseph form on 6 symmetric entries
    float innov = obs - (s0 + s2);
    float c00 = p00 + p02, c01 = p01 + p12, c02 = p02 + p22;
    float S = c00 + c02 + R_w;
    float invS = fast_rcpf(S);                // v_rcp_f32: off the div-fixup chain
    float K0 = c00 * invS, K1 = c01 * invS, K2 = c02 * invS;
    s0 = fmaf(K0, innov, s0);
    s1 = fmaf(K1, innov, s1);
    s2 = fmaf(K2, innov, s2);
    float sr = S - R_w;
    float d0 = c00 - K0 * sr, d1 = c01 - K1 * sr, d2 = c02 - K2 * sr;
    float m00 = p00 - K0 * c00, m01 = p01 - K0 * c01, m02 = p02 - K0 * c02;
    float m11 = p11 - K1 * c01, m12 = p12 - K1 * c02, m22 = p22 - K2 * c02;
    p00 = m00 - K0 * d0 + R_w * K0 * K0;
    p01 = m01 - K1 * d0 + R_w * K0 * K1;
    p02 = m02 - K2 * d0 + R_w * K0 * K2;
    p11 = m11 - K1 * d1 + R_w * K1 * K1;
    p12 = m12 - K2 * d1 + R_w * K1 * K2;
    p22 = m22 - K2 * d2 + R_w * K2 * K2;
  }

  // ---------------- forecast: HF prediction-only steps ----------------
  const float* rowVF  = A.v_fut  + (size_t)e * HFUT;
  const float* rowDTF = A.dt_fut + (size_t)e * HFUT;
  float* outY  = A.out + (size_t)e * HFUT;
  float* outYV = A.out + (size_t)BATCH * HFUT + (size_t)e * HFUT;
  float vprev = rowV[LHIST - 1];
#pragma unroll 1
  for (int t = 0; t < HFUT; ++t) {
    float dtk = rowDTF[t];
    float vcu = rowVF[t];
    float dv  = vcu - vprev;
    vprev = vcu;
    predict(dtk, vcu, dv);
    if (lo) {                                 // co-owners: low half stores
      outY[t]  = s0 + s2;
      outYV[t] = p00 + 2.0f * p02 + p22 + R_w;
    }
  }
  if (lo) A.out[2 * (size_t)BATCH * HFUT + e] = s1;   // final s[:,1]
}

extern "C" void kernel_launch(void* const* d_in, const int* in_sizes, int n_in,
                              void* d_out, int out_size, void* d_ws, size_t ws_size,
                              hipStream_t stream) {
  (void)in_sizes; (void)n_in; (void)out_size; (void)d_ws; (void)ws_size;
  KArgs a;
  a.v_hist      = (const float*)d_in[0];
  a.dt_hist     = (const float*)d_in[1];
  a.x_obs       = (const float*)d_in[2];
  a.v_fut       = (const float*)d_in[3];
  a.dt_fut      = (const float*)d_in[4];
  a.log_tau_n   = (const float*)d_in[5];
  a.log_q_n     = (const float*)d_in[6];
  a.log_R_white = (const float*)d_in[7];
  a.log_P0_nn   = (const float*)d_in[8];
  a.log_p0_xx   = (const float*)d_in[9];
  a.log_p0_uu   = (const float*)d_in[10];
  a.alpha       = (const float*)d_in[11];
  a.c           = (const float*)d_in[12];
  a.kappa       = (const float*)d_in[13];
  a.vc          = (const float*)d_in[14];
  a.qx          = (const float*)d_in[15];
  a.qu          = (const float*)d_in[16];
  a.q_scale     = (const float*)d_in[17];
  a.W1          = (const float*)d_in[18];
  a.b1          = (const float*)d_in[19];
  a.W2          = (const float*)d_in[20];
  a.b2          = (const float*)d_in[21];
  a.out         = (float*)d_out;

  dim3 grid(BATCH / 16), block(32);           // 512 waves, wave32, EXEC all-ones
  kf_neural_closure_kernel<<<grid, block, 0, stream>>>(a);
}